// VCSMC_30777735643644
// MI455X (gfx1250) — compile-verified
//
#include <hip/hip_runtime.h>
#include <math.h>

// ---------------- problem constants (from reference) ----------------
constexpr int Kp     = 256;   // particles
constexpr int Nt     = 32;    // taxa
constexpr int Sn     = 512;   // batched sites
constexpr int Sfull  = 1024;  // full sites
constexpr int An     = 4;     // alphabet
constexpr int Dn     = 64;    // embedding dim
constexpr int Cn     = 16;    // site feature dim
constexpr int Rounds = Nt - 1;            // 31
constexpr int FelRows = Nt + Kp * Rounds; // 7968 rows of S*A floats
constexpr float PB = 0.1f;                // PRIOR_B

typedef __attribute__((ext_vector_type(2))) float v2f;
typedef __attribute__((ext_vector_type(8))) float v8f;

// Full-precision fp32 WMMA: D = A(16x4) * B(4x16) + C  (matrix pipe, no downcast)
__device__ __forceinline__ v8f wmma4(v2f a, v2f b, v8f c) {
  return __builtin_amdgcn_wmma_f32_16x16x4_f32(false, a, false, b, (short)0, c,
                                               false, false);
}

__device__ __forceinline__ float lse4(float a, float b, float c, float d) {
  float m = fmaxf(fmaxf(a, b), fmaxf(c, d));
  return m + logf(expf(a - m) + expf(b - m) + expf(c - m) + expf(d - m));
}

__device__ __forceinline__ float softplus(float x) {
  return fmaxf(x, 0.f) + log1pf(expf(-fabsf(x)));
}

__device__ __forceinline__ float hash01(unsigned r, unsigned k) {
  unsigned long long z = (((unsigned long long)r) << 32) ^ (unsigned long long)k;
  z += 0x9E3779B97F4A7C15ull;
  z = (z ^ (z >> 30)) * 0xBF58476D1CE4E5B9ull;
  z = (z ^ (z >> 27)) * 0x94D049BB133111EBull;
  z = z ^ (z >> 31);
  return (float)(z >> 40) * (1.0f / 16777216.0f);
}

// ---------------- setup kernels ----------------

// site_SxC = site_positions(512x1024) @ W_site(1024x16), fp32 WMMA
__global__ __launch_bounds__(32) void k_site_gemm(
    const float* __restrict__ sp, const float* __restrict__ Wsite,
    float* __restrict__ siteC) {
  const int lane = threadIdx.x, half = lane >> 4, l16 = lane & 15;
  const int row = blockIdx.x * 16 + l16;
  const float* arow = sp + (size_t)row * Sfull;
  v8f acc = {0, 0, 0, 0, 0, 0, 0, 0};
  for (int kk = 0; kk < Sfull; kk += 4) {
    v2f a; a.x = arow[kk + 2 * half];          a.y = arow[kk + 2 * half + 1];
    v2f b; b.x = Wsite[(kk + 2 * half) * Cn + l16];
           b.y = Wsite[(kk + 2 * half + 1) * Cn + l16];
    acc = wmma4(a, b, acc);
  }
#pragma unroll
  for (int i = 0; i < 8; i++) {
    int m = blockIdx.x * 16 + i + 8 * half;
    siteC[m * Cn + l16] = acc[i];
  }
}

// site_logits(SxA) = siteC(SxC) @ W_stat_site(CxA)
__global__ void k_site_logits(const float* __restrict__ siteC,
                              const float* __restrict__ Wss,
                              float* __restrict__ siteL) {
  int i = blockIdx.x * blockDim.x + threadIdx.x;
  if (i >= Sn * An) return;
  int s = i >> 2, a = i & 3;
  float v = 0.f;
#pragma unroll
  for (int c = 0; c < Cn; c++) v += siteC[s * Cn + c] * Wss[c * An + a];
  siteL[i] = v;
}

// emb0(NxD) = mean_s(data) @ W_enc
__global__ void k_emb0(const float* __restrict__ data,
                       const float* __restrict__ Wenc,
                       float* __restrict__ emb0) {
  int n = blockIdx.x;
  __shared__ float mean[An];
  __shared__ float red[256];
  float acc[An] = {0, 0, 0, 0};
  for (int s = threadIdx.x; s < Sfull; s += blockDim.x)
#pragma unroll
    for (int a = 0; a < An; a++) acc[a] += data[((size_t)n * Sfull + s) * An + a];
#pragma unroll
  for (int a = 0; a < An; a++) {
    red[threadIdx.x] = acc[a]; __syncthreads();
    for (int o = 128; o > 0; o >>= 1) {
      if (threadIdx.x < o) red[threadIdx.x] += red[threadIdx.x + o];
      __syncthreads();
    }
    if (threadIdx.x == 0) mean[a] = red[0] / (float)Sfull;
    __syncthreads();
  }
  for (int d = threadIdx.x; d < Dn; d += blockDim.x) {
    float v = 0.f;
#pragma unroll
    for (int a = 0; a < An; a++) v += mean[a] * Wenc[a * Dn + d];
    emb0[n * Dn + d] = v;
  }
}

__global__ void k_init_state(const float* __restrict__ emb0, float* __restrict__ emb,
                             int* __restrict__ lc, int* __restrict__ slots,
                             float* __restrict__ bsum, float* __restrict__ logpi,
                             float* __restrict__ lw, float* __restrict__ logZ,
                             float* __restrict__ ldf) {
  int k = blockIdx.x;
  for (int x = threadIdx.x; x < Nt * Dn; x += blockDim.x) {
    int j = x / Dn, d = x % Dn;
    emb[((size_t)k * Nt + j) * Dn + d] = emb0[j * Dn + d];
  }
  for (int j = threadIdx.x; j < Nt; j += blockDim.x) {
    lc[k * Nt + j] = 1;
    slots[k * Nt + j] = j;  // leaf rows of fels pool, shared by all particles
  }
  if (threadIdx.x == 0) { bsum[k] = 0.f; logpi[k] = 0.f; lw[k] = -logf((float)Kp); }
  if (k == 0 && threadIdx.x == 0) {
    logZ[0] = 0.f;
    ldf[0] = 0.f; ldf[1] = 0.f;
    for (int n2 = 2; n2 <= 2 * Nt; n2++) ldf[n2] = ldf[n2 - 2] + logf((float)n2);
  }
}

__global__ void k_init_fels(const float* __restrict__ datab, float* __restrict__ fels) {
  int j = blockIdx.x;
  for (int x = threadIdx.x; x < Sn * An; x += blockDim.x)
    fels[(size_t)j * Sn * An + x] = logf(datab[(size_t)j * Sn * An + x]);
}

// ---------------- per-round kernels ----------------

// multinomial resampling via hash-uniform inverse-CDF (single wave-group block)
__global__ void k_resample(const float* __restrict__ lw, int* __restrict__ idx, int r) {
  __shared__ float sh[Kp];
  __shared__ float cdf[Kp];
  int k = threadIdx.x;
  sh[k] = lw[k]; __syncthreads();
  for (int o = Kp / 2; o > 0; o >>= 1) {
    if (k < o) sh[k] = fmaxf(sh[k], sh[k + o]);
    __syncthreads();
  }
  float m = sh[0]; __syncthreads();
  cdf[k] = expf(lw[k] - m); __syncthreads();
  for (int o = 1; o < Kp; o <<= 1) {            // Hillis-Steele inclusive scan
    float add = (k >= o) ? cdf[k - o] : 0.f;
    __syncthreads();
    cdf[k] += add;
    __syncthreads();
  }
  float u = hash01((unsigned)r, (unsigned)k) * cdf[Kp - 1];
  int lo = 0, hi = Kp - 1;
  while (lo < hi) { int mid = (lo + hi) >> 1; if (cdf[mid] < u) lo = mid + 1; else hi = mid; }
  idx[k] = lo;
}

// Gather parent state by slot indirection, build pair, compute branch lengths.
__global__ void k_gather(const int* __restrict__ idx,
                         const float* __restrict__ embO, float* __restrict__ embN,
                         const int* __restrict__ lcO, int* __restrict__ lcN,
                         const int* __restrict__ slO, int* __restrict__ slN,
                         const float* __restrict__ bsO, float* __restrict__ bsN,
                         const float* __restrict__ lpO, float* __restrict__ lpPrev,
                         float* __restrict__ pairB, const float* __restrict__ Wb,
                         float* __restrict__ b1, float* __restrict__ b2,
                         int* __restrict__ src01, float* __restrict__ llik,
                         int r, int t) {
  int k = blockIdx.x;
  int p = idx[k];
  int t2 = t - 1;
  __shared__ float sbr[2];
  // pair = concat(emb[p][0], emb[p][1])
  for (int d = threadIdx.x; d < 2 * Dn; d += blockDim.x)
    pairB[k * 2 * Dn + d] =
        embO[((size_t)p * Nt + (d < Dn ? 0 : 1)) * Dn + (d < Dn ? d : d - Dn)];
  // tail embeddings: new j=1..t2-1 <- old j+1
  for (int x = threadIdx.x; x < (t2 - 1) * Dn; x += blockDim.x) {
    int j = 1 + x / Dn, d = x % Dn;
    embN[((size_t)k * Nt + j) * Dn + d] = embO[((size_t)p * Nt + j + 1) * Dn + d];
  }
  if (threadIdx.x < Nt) {
    int j = threadIdx.x, lcv, slv;
    if (j == 0)       { lcv = lcO[p * Nt] + lcO[p * Nt + 1]; slv = Nt + r * Kp + k; }
    else if (j < t2)  { lcv = lcO[p * Nt + j + 1];           slv = slO[p * Nt + j + 1]; }
    else              { lcv = 1;                             slv = 0; }   // padding
    lcN[k * Nt + j] = lcv; slN[k * Nt + j] = slv;
  }
  if (threadIdx.x == 0) {
    src01[2 * k]     = slO[p * Nt];
    src01[2 * k + 1] = slO[p * Nt + 1];
    lpPrev[k] = lpO[p];
    llik[k] = 0.f;
  }
  __syncthreads();
  if (threadIdx.x < 2) {   // br = softplus(pair @ W_branch) + 1e-4
    float acc = 0.f;
    for (int i = 0; i < 2 * Dn; i++) acc += pairB[k * 2 * Dn + i] * Wb[i * 2 + threadIdx.x];
    sbr[threadIdx.x] = softplus(acc) + 1e-4f;
  }
  __syncthreads();
  if (threadIdx.x == 0) {
    b1[k] = sbr[0]; b2[k] = sbr[1];
    bsN[k] = bsO[p] + sbr[0] + sbr[1];
  }
}

// merged = pair(Kx128) @ W_merge(128x64)  -> embN[:,0,:]   (fp32 WMMA)
__global__ __launch_bounds__(32) void k_merge_gemm(
    const float* __restrict__ pairB, const float* __restrict__ Wm,
    float* __restrict__ embN) {
  const int lane = threadIdx.x, half = lane >> 4, l16 = lane & 15;
  const int rowA = blockIdx.x * 16 + l16;
  const int colBase = blockIdx.y * 16;
  const v2f* arow = (const v2f*)(pairB + (size_t)rowA * 2 * Dn);
  v8f acc = {0, 0, 0, 0, 0, 0, 0, 0};
  for (int kk = 0; kk < 2 * Dn; kk += 4) {
    v2f a = arow[(kk >> 1) + half];
    v2f b; b.x = Wm[(kk + 2 * half) * Dn + colBase + l16];
           b.y = Wm[(kk + 2 * half + 1) * Dn + colBase + l16];
    acc = wmma4(a, b, acc);
  }
#pragma unroll
  for (int i = 0; i < 8; i++) {
    int m = blockIdx.x * 16 + i + 8 * half;        // particle index
    embN[(size_t)m * Nt * Dn + colBase + l16] = acc[i];  // slot j=0
  }
}

// emb_logits(K*t2 x 4) = emb rows (Kx t2 of N slots, D) @ W_stat_emb(64x4), N padded to 16
__global__ __launch_bounds__(32) void k_stat_gemm(
    const float* __restrict__ embN, const float* __restrict__ Wse,
    float* __restrict__ embLog, int Mrows, int t2) {
  const int lane = threadIdx.x, half = lane >> 4, l16 = lane & 15;
  const int g = blockIdx.x * 16 + l16;       // logical row (always < Mrows: K*t2 %16==0)
  const int kpart = g / t2, j = g % t2;
  const float* arow = embN + ((size_t)kpart * Nt + j) * Dn;
  v8f acc = {0, 0, 0, 0, 0, 0, 0, 0};
  for (int kk = 0; kk < Dn; kk += 4) {
    v2f a; a.x = arow[kk + 2 * half]; a.y = arow[kk + 2 * half + 1];
    v2f b;
    b.x = (l16 < An) ? Wse[(kk + 2 * half) * An + l16] : 0.f;
    b.y = (l16 < An) ? Wse[(kk + 2 * half + 1) * An + l16] : 0.f;
    acc = wmma4(a, b, acc);
  }
#pragma unroll
  for (int i = 0; i < 8; i++) {
    int row = blockIdx.x * 16 + i + 8 * half;
    if (l16 < An && row < Mrows) embLog[row * An + l16] = acc[i];
  }
}

// Felsenstein update for the merged lineage: new fels row = c1 + c2
__global__ void k_c1c2(const float* __restrict__ embLog, const float* __restrict__ siteL,
                       const float* __restrict__ b1, const float* __restrict__ b2,
                       const int* __restrict__ src01, float* __restrict__ fels,
                       int r, int t2) {
  int k = blockIdx.x;
  int s = blockIdx.y * blockDim.x + threadIdx.x;
  const int g = k * t2;  // merged lineage row in embLog
  float l[An], pr[An];
#pragma unroll
  for (int a = 0; a < An; a++) l[a] = embLog[g * An + a] + siteL[s * An + a];
  float m = fmaxf(fmaxf(l[0], l[1]), fmaxf(l[2], l[3]));
  float lsev = m + logf(expf(l[0] - m) + expf(l[1] - m) + expf(l[2] - m) + expf(l[3] - m));
#pragma unroll
  for (int a = 0; a < An; a++) pr[a] = expf(l[a] - lsev);   // stat_m

  const float* f1 = fels + (size_t)src01[2 * k] * (Sn * An) + s * An;
  const float* f2 = fels + (size_t)src01[2 * k + 1] * (Sn * An) + s * An;
  float fa[An], fb[An];
#pragma unroll
  for (int a = 0; a < An; a++) { fa[a] = f1[a]; fb[a] = f2[a]; }

  float e1 = expf(-b1[k]), e2 = expf(-b2[k]);
  float off1[An], dia1[An], off2[An], dia2[An];
#pragma unroll
  for (int a = 0; a < An; a++) {
    off1[a] = logf((1.f - e1) * pr[a] + 1e-30f);
    dia1[a] = logf((1.f - e1) * pr[a] + e1 + 1e-30f);
    off2[a] = logf((1.f - e2) * pr[a] + 1e-30f);
    dia2[a] = logf((1.f - e2) * pr[a] + e2 + 1e-30f);
  }
  float* outrow = fels + (size_t)(Nt + r * Kp + k) * (Sn * An) + s * An;
#pragma unroll
  for (int i = 0; i < An; i++) {
    float t0 = ((0 == i) ? dia1[0] : off1[0]) + fa[0];
    float t1 = ((1 == i) ? dia1[1] : off1[1]) + fa[1];
    float t2v = ((2 == i) ? dia1[2] : off1[2]) + fa[2];
    float t3 = ((3 == i) ? dia1[3] : off1[3]) + fa[3];
    float c1 = lse4(t0, t1, t2v, t3);
    float u0 = ((0 == i) ? dia2[0] : off2[0]) + fb[0];
    float u1 = ((1 == i) ? dia2[1] : off2[1]) + fb[1];
    float u2 = ((2 == i) ? dia2[2] : off2[2]) + fb[2];
    float u3 = ((3 == i) ? dia2[3] : off2[3]) + fb[3];
    float c2 = lse4(u0, u1, u2, u3);
    outrow[i] = c1 + c2;
  }
}

// log_lik[k] = sum_{j,s} lse_a( log_softmax(logits)_a + fels_a )
__global__ void k_loglik(const float* __restrict__ embLog, const float* __restrict__ siteL,
                         const float* __restrict__ fels, const int* __restrict__ slots,
                         float* __restrict__ llik, int t2) {
  int g = blockIdx.x;
  int k = g / t2, j = g % t2;
  float e[An];
#pragma unroll
  for (int a = 0; a < An; a++) e[a] = embLog[g * An + a];
  const float* frow = fels + (size_t)slots[k * Nt + j] * (Sn * An);
  float acc = 0.f;
  for (int s = threadIdx.x; s < Sn; s += blockDim.x) {
    float l0 = e[0] + siteL[s * An + 0], l1 = e[1] + siteL[s * An + 1];
    float l2 = e[2] + siteL[s * An + 2], l3 = e[3] + siteL[s * An + 3];
    float m = fmaxf(fmaxf(l0, l1), fmaxf(l2, l3));
    float lsev = m + logf(expf(l0 - m) + expf(l1 - m) + expf(l2 - m) + expf(l3 - m));
    acc += lse4(l0 - lsev + frow[s * An + 0], l1 - lsev + frow[s * An + 1],
                l2 - lsev + frow[s * An + 2], l3 - lsev + frow[s * An + 3]);
  }
  __shared__ float red[256];
  red[threadIdx.x] = acc; __syncthreads();
  for (int o = 128; o > 0; o >>= 1) {
    if (threadIdx.x < o) red[threadIdx.x] += red[threadIdx.x + o];
    __syncthreads();
  }
  if (threadIdx.x == 0) atomicAdd(&llik[k], red[0]);
}

// weights, log_pi, logsumexp -> logZ accumulator (single block of K threads)
__global__ void k_weight(const float* __restrict__ llik, const float* __restrict__ bsum,
                         const int* __restrict__ lc, const float* __restrict__ lpPrev,
                         float* __restrict__ lpNew, float* __restrict__ lw,
                         float* __restrict__ logZ, const float* __restrict__ ldf,
                         int r, int t, int t2) {
  int k = threadIdx.x;
  float bp = -2.f * (float)(r + 1) * logf(PB) - bsum[k] / PB;
  float topo = 0.f; int nm = 0;
  for (int j = 0; j < t2; j++) {
    int c = lc[k * Nt + j];
    if (c > 1) { topo -= ldf[2 * c - 3]; nm++; }
  }
  float lvm = logf((float)(nm > 0 ? nm : 1));
  float lvp = logf((float)(t * (t - 1)) * 0.5f);
  float lpi = llik[k] + bp + topo;
  lpNew[k] = lpi;
  float w = lpi - lpPrev[k] + lvm - lvp;
  lw[k] = w;
  __shared__ float sh[Kp];
  sh[k] = w; __syncthreads();
  for (int o = Kp / 2; o > 0; o >>= 1) {
    if (k < o) sh[k] = fmaxf(sh[k], sh[k + o]);
    __syncthreads();
  }
  float m = sh[0]; __syncthreads();
  sh[k] = expf(w - m); __syncthreads();
  for (int o = Kp / 2; o > 0; o >>= 1) {
    if (k < o) sh[k] += sh[k + o];
    __syncthreads();
  }
  if (k == 0) logZ[0] += m + logf(sh[0]) - logf((float)Kp);
}

__global__ void k_final(const float* __restrict__ logZ, const float* __restrict__ llik,
                        float* __restrict__ out) {
  int k = threadIdx.x;
  if (k == 0) out[0] = logZ[0];
  out[1 + k] = llik[k];
}

// ---------------- host side ----------------
extern "C" void kernel_launch(void* const* d_in, const int* in_sizes, int n_in,
                              void* d_out, int out_size, void* d_ws, size_t ws_size,
                              hipStream_t stream) {
  (void)in_sizes; (void)n_in; (void)out_size; (void)ws_size;
  const float* data  = (const float*)d_in[0];  // N x Sfull x A
  const float* datab = (const float*)d_in[1];  // N x S x A
  const float* sp    = (const float*)d_in[2];  // S x Sfull
  const float* Wenc  = (const float*)d_in[3];  // A x D
  const float* Wsite = (const float*)d_in[4];  // Sfull x C
  const float* Wse   = (const float*)d_in[5];  // D x A
  const float* Wss   = (const float*)d_in[6];  // C x A
  const float* Wm    = (const float*)d_in[7];  // 2D x D
  const float* Wb    = (const float*)d_in[8];  // 2D x 2
  float* out = (float*)d_out;

  float* ws = (float*)d_ws;
  size_t off = 0;
  float* fels   = ws + off; off += (size_t)FelRows * Sn * An;  // 65.3 MB
  float* siteC  = ws + off; off += (size_t)Sn * Cn;
  float* siteL  = ws + off; off += (size_t)Sn * An;
  float* emb0   = ws + off; off += (size_t)Nt * Dn;
  float* embA   = ws + off; off += (size_t)Kp * Nt * Dn;
  float* embB   = ws + off; off += (size_t)Kp * Nt * Dn;
  float* pairB  = ws + off; off += (size_t)Kp * 2 * Dn;
  float* embLog = ws + off; off += (size_t)Kp * Rounds * An;
  float* b1     = ws + off; off += Kp;
  float* b2     = ws + off; off += Kp;
  float* bsumA  = ws + off; off += Kp;
  float* bsumB  = ws + off; off += Kp;
  float* lpiA   = ws + off; off += Kp;
  float* lpiB   = ws + off; off += Kp;
  float* lpPrev = ws + off; off += Kp;
  float* lw     = ws + off; off += Kp;
  float* llik   = ws + off; off += Kp;
  float* logZ   = ws + off; off += 8;
  float* ldf    = ws + off; off += 2 * Nt + 8;
  int* lcA    = (int*)(ws + off); off += (size_t)Kp * Nt;
  int* lcB    = (int*)(ws + off); off += (size_t)Kp * Nt;
  int* slotsA = (int*)(ws + off); off += (size_t)Kp * Nt;
  int* slotsB = (int*)(ws + off); off += (size_t)Kp * Nt;
  int* idx    = (int*)(ws + off); off += Kp;
  int* src01  = (int*)(ws + off); off += 2 * Kp;

  // setup (re-done every call: ws is not preserved between calls)
  k_site_gemm  <<<Sn / 16, 32, 0, stream>>>(sp, Wsite, siteC);
  k_site_logits<<<(Sn * An + 255) / 256, 256, 0, stream>>>(siteC, Wss, siteL);
  k_emb0       <<<Nt, 256, 0, stream>>>(data, Wenc, emb0);
  k_init_state <<<Kp, 128, 0, stream>>>(emb0, embA, lcA, slotsA, bsumA, lpiA, lw, logZ, ldf);
  k_init_fels  <<<Nt, 256, 0, stream>>>(datab, fels);

  float *embO = embA, *embN = embB, *bsO = bsumA, *bsN = bsumB;
  float *lpO = lpiA, *lpN = lpiB;
  int *lcO = lcA, *lcN = lcB, *slO = slotsA, *slN = slotsB;

  for (int r = 0; r < Rounds; r++) {
    const int t = Nt - r, t2 = t - 1;
    const int Mrows = Kp * t2;  // multiple of 16
    k_resample  <<<1, Kp, 0, stream>>>(lw, idx, r);
    k_gather    <<<Kp, 128, 0, stream>>>(idx, embO, embN, lcO, lcN, slO, slN,
                                         bsO, bsN, lpO, lpPrev, pairB, Wb,
                                         b1, b2, src01, llik, r, t);
    k_merge_gemm<<<dim3(Kp / 16, Dn / 16), 32, 0, stream>>>(pairB, Wm, embN);
    k_stat_gemm <<<Mrows / 16, 32, 0, stream>>>(embN, Wse, embLog, Mrows, t2);
    k_c1c2      <<<dim3(Kp, Sn / 256), 256, 0, stream>>>(embLog, siteL, b1, b2,
                                                         src01, fels, r, t2);
    k_loglik    <<<Mrows, 256, 0, stream>>>(embLog, siteL, fels, slN, llik, t2);
    k_weight    <<<1, Kp, 0, stream>>>(llik, bsN, lcN, lpPrev, lpN, lw, logZ, ldf, r, t, t2);
    // swap double buffers
    float* tf;
    tf = embO; embO = embN; embN = tf;
    tf = bsO;  bsO = bsN;   bsN = tf;
    tf = lpO;  lpO = lpN;   lpN = tf;
    int* ti;
    ti = lcO; lcO = lcN; lcN = ti;
    ti = slO; slO = slN; slN = ti;
  }
  k_final<<<1, Kp, 0, stream>>>(logZ, llik, out);
}